// RoI2Det_56040733278322
// MI455X (gfx1250) — compile-verified
//
#include <hip/hip_runtime.h>
#include <hip/hip_bf16.h>

typedef __attribute__((ext_vector_type(2))) float v2f;
typedef __attribute__((ext_vector_type(8))) float v8f;

#define NUM_CLASSES 80
#define NCOL 81          // classes + background
#define KPAD 84          // 81 padded to multiple of 4 for WMMA K-steps
#define NPROP 8192
#define NIMG 4
#define CAP 32768        // per-image raw candidate capacity
#define PRE_NMS_K 2048
#define SEL_CAP 4096     // compacted candidates entering the bitonic sort
#define MAX_OUT 100
#define KEEP_CAP 112     // >100 kept per class can never reach global top-100
#define SCORE_THR 0.05f
#define IOU_THR 0.5f
#define MAX_RATIO 4.135166556742356f  // |log(16/1000)|

// ---------------------------------------------------------------------------
// Kernel 0: zero per-image candidate counters (must rerun every launch)
// ---------------------------------------------------------------------------
__global__ void k_init(int* cnt) {
    if (threadIdx.x < NIMG) cnt[threadIdx.x] = 0;
}

// ---------------------------------------------------------------------------
// Kernel 1: softmax + threshold. Row sums of exp(logits) computed on the
// matrix pipe: V_WMMA_F32_16X16X4_F32 with a ones B-matrix reduces 16 rows
// x 4 classes per step; exp (TRANS) co-executes with WMMA (XDL) on CDNA5.
// A-layout (32-bit 16x4): lanes 0-15 hold M=0..15 {K0,K1}; lanes 16-31 {K2,K3}.
// D-layout: VGPR r, lanes 0-15 -> row r, lanes 16-31 -> row 8+r (all N equal).
// ---------------------------------------------------------------------------
__global__ void k_scores(const float* __restrict__ cls,
                         int* cnt, float* cscore, int* cidx) {
    const int wave = threadIdx.x >> 5;
    const int lane = threadIdx.x & 31;
    const int m    = lane & 15;
    const int half = lane >> 4;
    const int img  = blockIdx.y;
    const int r0   = blockIdx.x * 128 + wave * 16;  // 8 waves x 16 rows

    __shared__ float expE[8][16][KPAD];
    __shared__ float invs[8][16];

    const float* rowp = cls + (size_t)(img * NPROP + r0 + m) * NCOL;
    __builtin_prefetch(rowp, 0, 0);
    __builtin_prefetch(rowp + 64, 0, 0);

    v2f ones; ones[0] = 1.0f; ones[1] = 1.0f;
    v8f acc = {0.f, 0.f, 0.f, 0.f, 0.f, 0.f, 0.f, 0.f};

    for (int s = 0; s < KPAD / 4; ++s) {
        const int c0 = 4 * s + 2 * half;
        float e0 = (c0     < NCOL) ? __expf(rowp[c0])     : 0.0f;
        float e1 = (c0 + 1 < NCOL) ? __expf(rowp[c0 + 1]) : 0.0f;
        expE[wave][m][c0]     = e0;
        expE[wave][m][c0 + 1] = e1;
        v2f a; a[0] = e0; a[1] = e1;
        acc = __builtin_amdgcn_wmma_f32_16x16x4_f32(
            false, a, false, ones, (short)0, acc, false, false);
    }

    if (lane == 0) {
#pragma unroll
        for (int r = 0; r < 8; ++r) invs[wave][r] = 1.0f / acc[r];
    }
    if (lane == 16) {
#pragma unroll
        for (int r = 0; r < 8; ++r) invs[wave][8 + r] = 1.0f / acc[r];
    }
    __syncthreads();

    for (int t = lane; t < 16 * NUM_CLASSES; t += 32) {
        const int r = t / NUM_CLASSES;
        const int c = t - r * NUM_CLASSES;
        const float p = expE[wave][r][c] * invs[wave][r];
        if (p > SCORE_THR) {
            const int pos = atomicAdd(cnt + img, 1);
            if (pos < CAP) {
                cscore[img * CAP + pos] = p;
                cidx[img * CAP + pos]   = (r0 + r) * NUM_CLASSES + c;
            }
        }
    }
}

// ---------------------------------------------------------------------------
// Kernel 2: exact top-2048 per image. Histogram on float bits -> cut bucket,
// compact survivors (<=4096), bitonic sort in LDS on composite key
// (scoreBits<<32)|~idx  => descending score, ascending index tie-break
// (matches jax.lax.top_k).
// ---------------------------------------------------------------------------
__device__ __forceinline__ int score_bucket(unsigned bits) {
    int b = (int)((bits - 0x3D400000u) >> 14);
    return b < 0 ? 0 : (b > 4095 ? 4095 : b);
}

__global__ void k_select_sort(const int* cnt, const float* cscore, const int* cidx,
                              float* s2s, int* s2i) {
    const int img = blockIdx.x;
    const int tid = threadIdx.x;  // 1024 threads
    __shared__ int hist[4096];
    __shared__ unsigned long long keys[SEL_CAP];
    __shared__ int selT, selCnt;

    int n = cnt[img]; if (n > CAP) n = CAP;
    for (int i = tid; i < 4096; i += 1024) hist[i] = 0;
    if (tid == 0) selCnt = 0;
    __syncthreads();

    const float* sc = cscore + (size_t)img * CAP;
    const int*   ci = cidx   + (size_t)img * CAP;

    for (int i = tid; i < n; i += 1024)
        atomicAdd(&hist[score_bucket(__float_as_uint(sc[i]))], 1);
    __syncthreads();

    if (tid == 0) {
        int run = 0, T = -1;
        for (int b = 4095; b >= 0; --b) {
            run += hist[b];
            if (run >= PRE_NMS_K) { T = b; break; }
        }
        selT = T;
    }
    __syncthreads();
    const int T = selT;

    for (int i = tid; i < n; i += 1024) {
        const unsigned bits = __float_as_uint(sc[i]);
        if (T < 0 || score_bucket(bits) >= T) {
            const int pos = atomicAdd(&selCnt, 1);
            if (pos < SEL_CAP)
                keys[pos] = ((unsigned long long)bits << 32) |
                            (unsigned)(~(unsigned)ci[i]);
        }
    }
    __syncthreads();
    int M = selCnt; if (M > SEL_CAP) M = SEL_CAP;
    for (int i = tid; i < SEL_CAP; i += 1024)
        if (i >= M) keys[i] = 0ull;
    __syncthreads();

    // Bitonic sort, descending.
    for (int k = 2; k <= SEL_CAP; k <<= 1) {
        for (int j = k >> 1; j > 0; j >>= 1) {
            for (int t = tid; t < SEL_CAP; t += 1024) {
                const int l = t ^ j;
                if (l > t) {
                    const unsigned long long A = keys[t];
                    const unsigned long long B = keys[l];
                    const bool up = ((t & k) == 0);
                    const bool sw = up ? (A < B) : (A > B);
                    if (sw) { keys[t] = B; keys[l] = A; }
                }
            }
            __syncthreads();
        }
    }

    for (int j = tid; j < PRE_NMS_K; j += 1024) {
        const unsigned long long kk = keys[j];
        const int o = img * PRE_NMS_K + j;
        if (kk == 0ull) { s2s[o] = 0.0f; s2i[o] = -1; }
        else {
            s2s[o] = __uint_as_float((unsigned)(kk >> 32));
            s2i[o] = (int)(~(unsigned)kk);
        }
    }
}

// ---------------------------------------------------------------------------
// Kernel 3: delta2bbox decode for the 2048 survivors per image only.
// ---------------------------------------------------------------------------
__global__ void k_decode(const float* __restrict__ reg,
                         const float* __restrict__ props,
                         const int* imgH, const int* imgW,
                         const int* s2i,
                         float* cbox, int* clab, int* keep) {
    const int t = blockIdx.x * blockDim.x + threadIdx.x;
    if (t >= NIMG * PRE_NMS_K) return;
    const int img = t / PRE_NMS_K;
    keep[t] = 0;
    const int idx = s2i[t];
    if (idx < 0) {
        clab[t] = -1;
        cbox[t * 4 + 0] = 0.f; cbox[t * 4 + 1] = 0.f;
        cbox[t * 4 + 2] = 0.f; cbox[t * 4 + 3] = 0.f;
        return;
    }
    const float W = (float)imgW[0];
    const float H = (float)imgH[0];
    const int n = idx / NUM_CLASSES;
    const int c = idx - n * NUM_CLASSES;
    const float* p = props + (size_t)(img * NPROP + n) * 4;
    const float* d = reg + (size_t)(img * NPROP + n) * (NUM_CLASSES * 4) + c * 4;
    __builtin_prefetch(d, 0, 0);

    const float dx = d[0] * 0.1f;
    const float dy = d[1] * 0.1f;
    float dw = d[2] * 0.2f;
    float dh = d[3] * 0.2f;
    dw = fminf(fmaxf(dw, -MAX_RATIO), MAX_RATIO);
    dh = fminf(fmaxf(dh, -MAX_RATIO), MAX_RATIO);

    const float px = (p[0] + p[2]) * 0.5f;
    const float py = (p[1] + p[3]) * 0.5f;
    const float pw = p[2] - p[0];
    const float ph = p[3] - p[1];
    const float gx = px + pw * dx;
    const float gy = py + ph * dy;
    const float gw = pw * __expf(dw);
    const float gh = ph * __expf(dh);

    cbox[t * 4 + 0] = fminf(fmaxf(gx - gw * 0.5f, 0.f), W);
    cbox[t * 4 + 1] = fminf(fmaxf(gy - gh * 0.5f, 0.f), H);
    cbox[t * 4 + 2] = fminf(fmaxf(gx + gw * 0.5f, 0.f), W);
    cbox[t * 4 + 3] = fminf(fmaxf(gy + gh * 0.5f, 0.f), H);
    clab[t] = c;
}

// ---------------------------------------------------------------------------
// Kernel 4: greedy NMS, one wave per (class, image). Cross-class IoU is zero
// under the reference's class-offset trick, so classes are independent.
// Wave-uniform ballot control flow; kept boxes live in LDS.
// ---------------------------------------------------------------------------
__device__ __forceinline__ float iou_pair(float ax1, float ay1, float ax2, float ay2,
                                          float bx1, float by1, float bx2, float by2) {
    const float areaA = fmaxf(ax2 - ax1, 0.f) * fmaxf(ay2 - ay1, 0.f);
    const float areaB = fmaxf(bx2 - bx1, 0.f) * fmaxf(by2 - by1, 0.f);
    const float ix = fmaxf(fminf(ax2, bx2) - fmaxf(ax1, bx1), 0.f);
    const float iy = fmaxf(fminf(ay2, by2) - fmaxf(ay1, by1), 0.f);
    const float inter = ix * iy;
    return inter / fmaxf(areaA + areaB - inter, 1e-6f);
}

__global__ void k_nms(const float* s2s, const int* clab, const float* cbox,
                      int* keep) {
    const int img  = blockIdx.y;
    const int cls  = blockIdx.x;
    const int lane = threadIdx.x;  // 32 threads = 1 wave
    __shared__ float kb[KEEP_CAP][4];
    int kcnt = 0;
    const int baseI = img * PRE_NMS_K;

    for (int base = 0; base < PRE_NMS_K; base += 32) {
        const int i = baseI + base + lane;
        const bool match = (clab[i] == cls) && (s2s[i] > 0.0f);
        unsigned mask = __builtin_amdgcn_ballot_w32(match);  // uniform
        while (mask) {
            const int off = __builtin_ctz(mask);
            mask &= mask - 1;
            const int j = baseI + base + off;
            const float bx1 = cbox[j * 4 + 0];
            const float by1 = cbox[j * 4 + 1];
            const float bx2 = cbox[j * 4 + 2];
            const float by2 = cbox[j * 4 + 3];

            bool sup = false;
            const int nslots = (kcnt + 31) >> 5;  // <= 4
            for (int s = 0; s < nslots; ++s) {
                const int slot = lane + (s << 5);
                if (slot < kcnt) {
                    if (iou_pair(bx1, by1, bx2, by2,
                                 kb[slot][0], kb[slot][1],
                                 kb[slot][2], kb[slot][3]) > IOU_THR)
                        sup = true;
                }
            }
            if (__builtin_amdgcn_ballot_w32(sup) == 0u) {  // uniform: keep j
                if (lane == 0) keep[j] = 1;
                if (kcnt < KEEP_CAP) {
                    if (lane == 0) {
                        kb[kcnt][0] = bx1; kb[kcnt][1] = by1;
                        kb[kcnt][2] = bx2; kb[kcnt][3] = by2;
                    }
                    __syncthreads();  // single-wave block: LDS visibility fence
                }
                kcnt++;
            }
        }
    }
}

// ---------------------------------------------------------------------------
// Kernel 5: select first 100 kept (sorted order == score order) and emit
// outputs; defaults 0 / -1 elsewhere. Labels/inds written as int32.
// ---------------------------------------------------------------------------
__global__ void k_finalize(const float* s2s, const int* s2i, const int* clab,
                           const float* cbox, const int* keep, float* outF) {
    const int img = blockIdx.x;
    const int tid = threadIdx.x;  // 256
    __shared__ int psum[256];
    __shared__ int pref[257];
    const int base = img * PRE_NMS_K;

    int s = 0;
#pragma unroll
    for (int r = 0; r < 8; ++r) s += keep[base + tid * 8 + r];
    psum[tid] = s;
    __syncthreads();
    if (tid == 0) {
        int run = 0;
        for (int t = 0; t < 256; ++t) { pref[t] = run; run += psum[t]; }
        pref[256] = run;
    }
    __syncthreads();

    float* boxesOut  = outF;                                   // [4,100,4]
    float* scoresOut = outF + NIMG * MAX_OUT * 4;              // [4,100]
    int*   labelsOut = (int*)outF + NIMG * MAX_OUT * 5;        // [4,100]
    int*   indsOut   = (int*)outF + NIMG * MAX_OUT * 6;        // [4,100]

    int rank = pref[tid];
    for (int r = 0; r < 8; ++r) {
        const int i = base + tid * 8 + r;
        if (keep[i]) {
            if (rank < MAX_OUT) {
                const int o = img * MAX_OUT + rank;
                boxesOut[o * 4 + 0] = cbox[i * 4 + 0];
                boxesOut[o * 4 + 1] = cbox[i * 4 + 1];
                boxesOut[o * 4 + 2] = cbox[i * 4 + 2];
                boxesOut[o * 4 + 3] = cbox[i * 4 + 3];
                scoresOut[o] = s2s[i];
                labelsOut[o] = clab[i];
                indsOut[o]   = s2i[i];
            }
            rank++;
        }
    }
    int total = pref[256]; if (total > MAX_OUT) total = MAX_OUT;
    if (tid < MAX_OUT && tid >= total) {
        const int o = img * MAX_OUT + tid;
        boxesOut[o * 4 + 0] = 0.f; boxesOut[o * 4 + 1] = 0.f;
        boxesOut[o * 4 + 2] = 0.f; boxesOut[o * 4 + 3] = 0.f;
        scoresOut[o] = 0.f;
        labelsOut[o] = -1;
        indsOut[o]   = -1;
    }
}

// ---------------------------------------------------------------------------
extern "C" void kernel_launch(void* const* d_in, const int* in_sizes, int n_in,
                              void* d_out, int out_size, void* d_ws, size_t ws_size,
                              hipStream_t stream) {
    (void)in_sizes; (void)n_in; (void)out_size; (void)ws_size;
    const float* cls   = (const float*)d_in[0];  // [B*N, 81]
    const float* reg   = (const float*)d_in[1];  // [B*N, 320]
    const float* props = (const float*)d_in[2];  // [B, N, 4]
    const int*   imgH  = (const int*)d_in[3];
    const int*   imgW  = (const int*)d_in[4];

    char* w = (char*)d_ws;
    size_t o = 0;
    int*   cnt    = (int*)(w + o);   o += 64;
    float* cscore = (float*)(w + o); o += (size_t)NIMG * CAP * 4;
    int*   cidx   = (int*)(w + o);   o += (size_t)NIMG * CAP * 4;
    float* s2s    = (float*)(w + o); o += (size_t)NIMG * PRE_NMS_K * 4;
    int*   s2i    = (int*)(w + o);   o += (size_t)NIMG * PRE_NMS_K * 4;
    float* cbox   = (float*)(w + o); o += (size_t)NIMG * PRE_NMS_K * 16;
    int*   clab   = (int*)(w + o);   o += (size_t)NIMG * PRE_NMS_K * 4;
    int*   keepA  = (int*)(w + o);   o += (size_t)NIMG * PRE_NMS_K * 4;

    k_init<<<1, 64, 0, stream>>>(cnt);

    dim3 g1(NPROP / 128, NIMG);
    k_scores<<<g1, 256, 0, stream>>>(cls, cnt, cscore, cidx);

    k_select_sort<<<NIMG, 1024, 0, stream>>>(cnt, cscore, cidx, s2s, s2i);

    k_decode<<<(NIMG * PRE_NMS_K + 255) / 256, 256, 0, stream>>>(
        reg, props, imgH, imgW, s2i, cbox, clab, keepA);

    dim3 g4(NUM_CLASSES, NIMG);
    k_nms<<<g4, 32, 0, stream>>>(s2s, clab, cbox, keepA);

    k_finalize<<<NIMG, 256, 0, stream>>>(s2s, s2i, clab, cbox, keepA,
                                         (float*)d_out);
}